// GenerateDepthFeature_Pts_9543417332323
// MI455X (gfx1250) — compile-verified
//
#include <hip/hip_runtime.h>
#include <cstdint>

#define DDIM 256
#define PTS_PER_BLOCK 8
#define STEP (6.0f / 255.0f)

// ---------------------------------------------------------------------------
// Kernel 1: zero-fill the [D,H,W] accumulation surface (128-bit stores).
// Regular-temporal stores on purpose: we WANT these lines resident+dirty in
// the 192MB L2 so the subsequent 51.2M fp32 atomics all hit in L2.
// ---------------------------------------------------------------------------
__global__ __launch_bounds__(256) void zero_kernel(float4* __restrict__ out4, int n4) {
  int i = blockIdx.x * blockDim.x + threadIdx.x;
  const int stride = gridDim.x * blockDim.x;
  const float4 z = make_float4(0.f, 0.f, 0.f, 0.f);
  for (; i < n4; i += stride) out4[i] = z;
}

// ---------------------------------------------------------------------------
// Kernel 2: one wave32 per point. Block = 256 threads = 8 waves = 8 points.
// Wave 0 stages the block's 8 point-records (32 dwords) into LDS with a
// single async global->LDS copy (CDNA5 ASYNCcnt path), then each wave:
//   - computes 8 gaussian pdf samples per lane (v_exp_f32)
//   - wave-reduces sum of squares (wave32 shfl_xor)
//   - scatters one s_clause'd burst of 8 native global_atomic_add_f32
//     (no-return -> STOREcnt; SGPR base + 32-bit lane offsets)
// ---------------------------------------------------------------------------
__global__ __launch_bounds__(256) void depth_scatter_kernel(
    const int* __restrict__ pts_pos,     // [N,2] (u,v)
    const int* __restrict__ pts_depth,   // [N]
    const float* __restrict__ pts_conf,  // [N]
    const int* __restrict__ feat_h,      // [1]
    const int* __restrict__ feat_w,      // [1]
    float* __restrict__ out,             // [D,H,W]
    int N) {
  // LDS layout (dwords): [0..7]=conf bits, [8..15]=depth, [16..31]=(u,v) pairs
  __shared__ uint32_t s_pts[32];

  const int tid  = threadIdx.x;
  const int lane = tid & 31;
  const int wav  = tid >> 5;
  const int base = blockIdx.x * PTS_PER_BLOCK;

  if (tid < 32) {
    // Per-lane 64-bit source address: lanes 0-7 conf, 8-15 depth, 16-31 pos.
    const void* gp;
    if (lane < 8) {
      int i = base + lane;            if (i >= N) i = N - 1;
      gp = pts_conf + i;
    } else if (lane < 16) {
      int i = base + (lane - 8);      if (i >= N) i = N - 1;
      gp = pts_depth + i;
    } else {
      int i = base * 2 + (lane - 16); int lim = 2 * N - 1; if (i > lim) i = lim;
      gp = pts_pos + i;
    }
    const uint64_t ga = (uint64_t)(uintptr_t)gp;
    // Generic pointer to LDS: low 32 bits are the wave-relative LDS offset.
    const uint32_t ldsoff = (uint32_t)(uintptr_t)(&s_pts[lane]);
    asm volatile("global_load_async_to_lds_b32 %0, %1, off"
                 :
                 : "v"(ldsoff), "v"(ga)
                 : "memory");
    asm volatile("s_wait_asynccnt 0" ::: "memory");
  }
  __syncthreads();

  const int p = base + wav;
  if (p >= N) return;

  const float sigma = __uint_as_float(s_pts[wav]);
  const int   depth = (int)s_pts[8 + wav];
  const int   u     = (int)s_pts[16 + 2 * wav];
  const int   v     = (int)s_pts[16 + 2 * wav + 1];

  const int W  = *feat_w;
  const int H  = *feat_h;
  const int HW = H * W;                   // 180224
  const int lin = v * W + u;              // pixel column in [0, HW)

  // 1/(sigma*sqrt(2pi)) cancels under L2 normalization -> drop it.
  const float inv_s = __builtin_amdgcn_rcpf(sigma);

  float pr[8];
  float sum = 0.0f;
#pragma unroll
  for (int k = 0; k < 8; ++k) {
    const int d = (k << 5) + lane;        // depth bin, 0..255
    int m = d - depth;
    if (d > depth) m -= 1;                // analytic d_coors slice
    const float x = (float)m * STEP;
    const float z = x * inv_s;
    const float e = __expf(-0.5f * z * z);
    pr[k] = e;
    sum += e * e;
  }

  // wave32 butterfly reduction of sum of squares
#pragma unroll
  for (int off = 16; off > 0; off >>= 1)
    sum += __shfl_xor(sum, off, 32);

  // sum >= 1 always (the d==depth sample is exp(0)=1), clamp can't fire.
  const float rn = __builtin_amdgcn_rsqf(sum);

  // Wave-uniform base (readfirstlane'd into an SGPR pair) + 32-bit per-lane
  // byte offsets: off[k] = (k*32+lane)*HW*4, max ~184MB < 2^32.
  const uint64_t basep = (uint64_t)(uintptr_t)(out + (size_t)lin);
  const uint32_t hw4   = (uint32_t)HW * 4u;
  uint32_t offb[8];
  float    val[8];
#pragma unroll
  for (int k = 0; k < 8; ++k) {
    offb[k] = (uint32_t)((k << 5) + lane) * hw4;
    val[k]  = pr[k] * rn;
  }

  // One clause of 8 fire-and-forget f32 atomics (same clause type: atomic
  // without return). Native instruction guaranteed — no CAS-loop fallback.
  asm volatile(
      "s_clause 0x7\n\t"
      "global_atomic_add_f32 %1, %9, %0\n\t"
      "global_atomic_add_f32 %2, %10, %0\n\t"
      "global_atomic_add_f32 %3, %11, %0\n\t"
      "global_atomic_add_f32 %4, %12, %0\n\t"
      "global_atomic_add_f32 %5, %13, %0\n\t"
      "global_atomic_add_f32 %6, %14, %0\n\t"
      "global_atomic_add_f32 %7, %15, %0\n\t"
      "global_atomic_add_f32 %8, %16, %0"
      :
      : "s"(basep),
        "v"(offb[0]), "v"(offb[1]), "v"(offb[2]), "v"(offb[3]),
        "v"(offb[4]), "v"(offb[5]), "v"(offb[6]), "v"(offb[7]),
        "v"(val[0]), "v"(val[1]), "v"(val[2]), "v"(val[3]),
        "v"(val[4]), "v"(val[5]), "v"(val[6]), "v"(val[7])
      : "memory");
}

extern "C" void kernel_launch(void* const* d_in, const int* in_sizes, int n_in,
                              void* d_out, int out_size, void* d_ws, size_t ws_size,
                              hipStream_t stream) {
  const int*   pts_pos   = (const int*)d_in[0];    // [N,2]
  const int*   pts_depth = (const int*)d_in[1];    // [N,1]
  const float* pts_conf  = (const float*)d_in[2];  // [N,1]
  const int*   feat_h    = (const int*)d_in[3];    // [1]
  const int*   feat_w    = (const int*)d_in[4];    // [1]
  float*       out       = (float*)d_out;          // [D,H,W] f32

  const int N  = in_sizes[0] / 2;
  const int n4 = out_size / 4;                     // out_size divisible by 4

  int zblocks = (n4 + 255) / 256;
  if (zblocks > 16384) zblocks = 16384;            // grid-stride
  zero_kernel<<<zblocks, 256, 0, stream>>>((float4*)out, n4);

  const int sblocks = (N + PTS_PER_BLOCK - 1) / PTS_PER_BLOCK;
  depth_scatter_kernel<<<sblocks, 256, 0, stream>>>(
      pts_pos, pts_depth, pts_conf, feat_h, feat_w, out, N);
}